// RTRL_81106162418164
// MI455X (gfx1250) — compile-verified
//
#include <hip/hip_runtime.h>

#define DD   256
#define HH   256
#define BB   4
#define PP   65792            // D*H + H
#define DHW  65536            // D*H
#define NC4  16448            // P/4
#define BETA 0.9f
#define ESZ  67371008LL       // B*H*P

typedef __attribute__((ext_vector_type(2))) float v2f;
typedef __attribute__((ext_vector_type(4))) float v4f;
typedef __attribute__((ext_vector_type(8))) float v8f;

// ---------------------------------------------------------------------------
// Kernel 1: s = tanh(BETA*u + x@W + b) via V_WMMA_F32_16X16X4_F32 (exact f32).
// One block, 16 waves (wave32). Wave w computes output columns [16w, 16w+16).
// x (4x256) is zero-padded to 16x256 in LDS so A-fragment loads are uniform
// (EXEC must be all-1s at WMMA).
// A 16x4 f32 layout: lane L (0-15) = row M=L, VGPR0/1 = K=0/1; lanes 16-31: K=2/3.
// B 4x16  f32 layout: lane L (0-15) = col N=L, VGPR0/1 = K=0/1; lanes 16-31: K=2/3.
// C/D 16x16 f32: lanes 0-15 N=lane, VGPR v = row M=v; lanes 16-31: M=v+8.
// ---------------------------------------------------------------------------
__global__ __launch_bounds__(512) void rtrl_small_wmma(
    const float* __restrict__ x, const float* __restrict__ W,
    const float* __restrict__ bias, const float* __restrict__ u,
    float* __restrict__ s_out, float* __restrict__ a_out)
{
    __shared__ float xs[16 * DD];            // 16 KB, rows 4..15 zero
    for (int i = threadIdx.x; i < 16 * DD; i += 512) xs[i] = 0.0f;
    __syncthreads();
    for (int i = threadIdx.x; i < BB * DD; i += 512) xs[i] = x[i];
    __syncthreads();

    const int lane = threadIdx.x & 31;
    const int wave = threadIdx.x >> 5;       // 0..15 -> N tile
    const int n0   = wave << 4;
    const int half = lane >> 4;              // 0: K=0,1   1: K=2,3
    const int l15  = lane & 15;

    v8f acc = {};
    for (int k = 0; k < 64; ++k) {
        const int kk = 4 * k + 2 * half;
        v2f a, b;
        a.x = xs[l15 * DD + kk];
        a.y = xs[l15 * DD + kk + 1];
        b.x = W[(long long)kk       * HH + n0 + l15];
        b.y = W[(long long)(kk + 1) * HH + n0 + l15];
        acc = __builtin_amdgcn_wmma_f32_16x16x4_f32(
                  false, a, false, b, (short)0, acc, false, false);
    }

    if (half == 0) {                         // lanes 0-15 hold rows 0..7 in v0..7
        const int col = n0 + l15;
        #pragma unroll
        for (int m = 0; m < BB; ++m) {
            float un = BETA * u[m * HH + col] + acc[m] + bias[col];
            float s  = tanhf(un);
            s_out[m * HH + col] = s;
            a_out[m * HH + col] = 1.0f - s * s;
        }
    }
}

// ---------------------------------------------------------------------------
// Kernel 2: streaming update over E (810 MB traffic, bandwidth bound).
//   E_new = BETA*E + du_dp  ;  ds_dtheta = (1-s^2) * E_new
// du_dp[h, d*H + h'] = x[i,d] iff h'==h ;  du_dp[h, DH + h'] = 1 iff h'==h.
// Non-temporal B128 accesses: working set (270MB in + 540MB out) >> 192MB L2.
// ---------------------------------------------------------------------------
__global__ __launch_bounds__(256) void rtrl_stream(
    const float* __restrict__ E, const float* __restrict__ x,
    const float* __restrict__ a_in,
    float* __restrict__ Enew, float* __restrict__ dsdt)
{
    const int row = blockIdx.y;              // i*H + h
    const int i   = row >> 8;
    const int h   = row & 255;
    const int c   = blockIdx.x * 256 + threadIdx.x;
    if (c >= NC4) return;

    const int p0 = c << 2;
    const long long base = (long long)row * PP + p0;

    v4f e = __builtin_nontemporal_load((const v4f*)(E + base));
    const float av = a_in[row];

    #pragma unroll
    for (int j = 0; j < 4; ++j) {
        const int p = p0 + j;
        float add = 0.0f;
        if (p < DHW) {
            if ((p & 255) == h) add = x[i * DD + (p >> 8)];
        } else {
            if ((p - DHW) == h) add = 1.0f;
        }
        e[j] = BETA * e[j] + add;
    }

    v4f ds;
    ds[0] = av * e[0]; ds[1] = av * e[1];
    ds[2] = av * e[2]; ds[3] = av * e[3];

    __builtin_nontemporal_store(e,  (v4f*)(Enew + base));
    __builtin_nontemporal_store(ds, (v4f*)(dsdt + base));
}

// ---------------------------------------------------------------------------
extern "C" void kernel_launch(void* const* d_in, const int* in_sizes, int n_in,
                              void* d_out, int out_size, void* d_ws, size_t ws_size,
                              hipStream_t stream)
{
    const float* x = (const float*)d_in[0];   // (4,256)
    const float* W = (const float*)d_in[1];   // (256,256)
    const float* b = (const float*)d_in[2];   // (256,)
    const float* u = (const float*)d_in[3];   // (4,256)
    const float* E = (const float*)d_in[4];   // (4,256,65792)

    float* s_out = (float*)d_out;             // 1024
    float* Enew  = s_out + (BB * HH);         // 67,371,008
    float* dsdt  = Enew + ESZ;                // 67,371,008
    float* a_ws  = (float*)d_ws;              // 1024 floats of (1-s^2)

    rtrl_small_wmma<<<1, 512, 0, stream>>>(x, W, b, u, s_out, a_ws);

    dim3 grid((NC4 + 255) / 256, BB * HH);    // (65, 1024)
    rtrl_stream<<<grid, 256, 0, stream>>>(E, x, a_ws, Enew, dsdt);
}